// GraphSparseNodeOnly_89275190215163
// MI455X (gfx1250) — compile-verified
//
#include <hip/hip_runtime.h>

#define N_NODES   100000
#define N_EDGES   3200000
#define N_GRAPHS  1024
#define D         128
#define FC1_DIM   64
#define FC2_DIM   10

typedef __attribute__((ext_vector_type(2))) float v2f;
typedef __attribute__((ext_vector_type(8))) float v8f;
typedef __attribute__((ext_vector_type(4))) unsigned int tdm_v4u;
typedef __attribute__((ext_vector_type(8))) int tdm_v8i;
typedef __attribute__((ext_vector_type(4))) int tdm_v4i;

// ---------------------------------------------------------------------------
// Zero-fill (accumulators live in ws, which the harness poisons with 0xAA)
// ---------------------------------------------------------------------------
__global__ void zero_f32(float* __restrict__ p, int n) {
  int i = blockIdx.x * blockDim.x + threadIdx.x;
  int stride = gridDim.x * blockDim.x;
  for (; i < n; i += stride) p[i] = 0.0f;
}

// ---------------------------------------------------------------------------
// Edge scatter: agg[dst[e]] += x[src[e]]   (one wave32 per edge)
// lane l loads float4 -> fully coalesced 512B gather per wave, 4 f32 atomics
// per lane. x rows (51MB) + agg (51MB) are L2-resident (192MB L2).
// ---------------------------------------------------------------------------
__global__ __launch_bounds__(256) void edge_scatter(
    const float* __restrict__ x, const int* __restrict__ src,
    const int* __restrict__ dst, float* __restrict__ agg, int nEdges) {
  unsigned gtid = blockIdx.x * blockDim.x + threadIdx.x;
  int edge = (int)(gtid >> 5);
  int lane = threadIdx.x & 31;
  if (edge >= nEdges) return;
  int s = __builtin_amdgcn_readfirstlane(src[edge]);
  int d = __builtin_amdgcn_readfirstlane(dst[edge]);
  const float4 v = ((const float4*)(x + (size_t)s * D))[lane];
  float* p = agg + (size_t)d * D + lane * 4;
  atomicAdd(p + 0, v.x);
  atomicAdd(p + 1, v.y);
  atomicAdd(p + 2, v.z);
  atomicAdd(p + 3, v.w);
}

// ---------------------------------------------------------------------------
// x_out = relu(agg @ W + b), fp32 via V_WMMA_F32_16X16X4_F32.
// * W (128x128 f32 = 64KB) staged into LDS once per block by the TDM
//   (tensor_load_to_lds, 2D descriptor), B fragments then come from LDS.
// * One wave owns a 16-row tile and ALL 8 column tiles: A fragment loaded
//   once per K-step, 8 WMMAs issued against it (8x A-traffic reduction).
// A (16x4 f32): lane = M, lane>>4 selects K pair; C/D: M = g + 8*(lane>>4).
// ---------------------------------------------------------------------------
__global__ __launch_bounds__(256) void conv_gemm_relu(
    const float* __restrict__ A, const float* __restrict__ W,
    const float* __restrict__ bias, float* __restrict__ out, int nRowTiles) {
  __shared__ float sW[D * D];   // 64 KB

  // ---- TDM: DMA the whole layer weight into LDS (issued by wave 0) ----
  if (threadIdx.x < 32) {
    const uint32_t ldsOff = (uint32_t)(uintptr_t)(&sW[0]);  // low 32b = LDS offset
    const uint64_t ga = (uint64_t)(uintptr_t)W;
    tdm_v4u g0;
    g0[0] = 1u;                                             // count=1 (valid D#)
    g0[1] = ldsOff;                                         // lds_addr (bytes)
    g0[2] = (uint32_t)ga;                                   // global_addr[31:0]
    g0[3] = ((uint32_t)(ga >> 32) & 0x01FFFFFFu)            // global_addr[56:32]
            | (2u << 30);                                   // type = 2 ("image")
    tdm_v8i g1;
    g1[0] = (2 << 16);       // data_size = 2 -> 4-byte elements; no multicast
    g1[1] = (D << 16);       // tensor_dim0 = 128 (bits 79:48, low half)
    g1[2] = (D << 16);       // tensor_dim1 = 128 (bits 111:80, low half)
    g1[3] = (D << 16);       // tile_dim0   = 128 (bits 127:112)
    g1[4] = D;               // tile_dim1   = 128 (bits 143:128)
    g1[5] = D;               // tensor_dim0_stride = 128 (bits 207:160, low)
    g1[6] = 0;
    g1[7] = 0;
    tdm_v4i gz4 = {0, 0, 0, 0};                             // 2D: groups 2/3 unused
    tdm_v8i gz8 = {0, 0, 0, 0, 0, 0, 0, 0};
    __builtin_amdgcn_tensor_load_to_lds(g0, g1, gz4, gz4, gz8, 0);
    __builtin_amdgcn_s_wait_tensorcnt(0);
  }
  __syncthreads();

  const int lane    = threadIdx.x & 31;
  const int rowTile = blockIdx.x * 8 + (threadIdx.x >> 5);
  if (rowTile >= nRowTiles) return;

  const int m     = lane & 15;          // M for A-frag, N for B-frag
  const int khalf = lane >> 4;          // K pair select
  const size_t rowA = (size_t)(rowTile * 16 + m) * D;

  v8f acc[8] = {};
  for (int k0 = 0; k0 < D; k0 += 4) {
    const int k = k0 + 2 * khalf;
    v2f a;
    a.x = A[rowA + k];
    a.y = A[rowA + k + 1];
    const float* wk = &sW[(size_t)k * D + m];
    #pragma unroll
    for (int ct = 0; ct < 8; ++ct) {
      v2f b;
      b.x = wk[ct * 16];                // W[k][ct*16+m]
      b.y = wk[ct * 16 + D];            // W[k+1][ct*16+m]
      acc[ct] = __builtin_amdgcn_wmma_f32_16x16x4_f32(false, a, false, b,
                                                      (short)0, acc[ct],
                                                      false, false);
    }
  }

  #pragma unroll
  for (int ct = 0; ct < 8; ++ct) {
    const float bv = bias[ct * 16 + m];
    #pragma unroll
    for (int g = 0; g < 8; ++g) {
      const int r = rowTile * 16 + g + 8 * khalf;
      float v = acc[ct][g] + bv;
      v = fmaxf(v, 0.0f);
      out[(size_t)r * D + ct * 16 + m] = v;
    }
  }
}

// ---------------------------------------------------------------------------
// Global sum pool: g[batching[node]] += x[node]   (one thread per element)
// ---------------------------------------------------------------------------
__global__ __launch_bounds__(256) void pool_scatter(
    const float* __restrict__ x, const int* __restrict__ batching,
    float* __restrict__ g) {
  int idx = blockIdx.x * blockDim.x + threadIdx.x;
  if (idx >= N_NODES * D) return;
  int node = idx >> 7;
  int f    = idx & (D - 1);
  int gi   = batching[node];
  atomicAdd(&g[(size_t)gi * D + f], x[idx]);
}

// ---------------------------------------------------------------------------
// Fused head: h = softmax((g @ Wf1 + bf1) @ Wf2 + bf2). One block per graph.
// ---------------------------------------------------------------------------
__global__ __launch_bounds__(64) void head_kernel(
    const float* __restrict__ g,
    const float* __restrict__ Wf1, const float* __restrict__ bf1,
    const float* __restrict__ Wf2, const float* __restrict__ bf2,
    float* __restrict__ out) {
  __shared__ float sg[D];
  __shared__ float sh1[FC1_DIM];
  __shared__ float sh2[FC2_DIM];
  __shared__ float red[2];
  const int gid = blockIdx.x;
  const int t   = threadIdx.x;

  sg[t]      = g[(size_t)gid * D + t];
  sg[t + 64] = g[(size_t)gid * D + t + 64];
  __syncthreads();

  float acc = bf1[t];
  #pragma unroll 4
  for (int k = 0; k < D; ++k) acc += sg[k] * Wf1[k * FC1_DIM + t];
  sh1[t] = acc;
  __syncthreads();

  if (t < FC2_DIM) {
    float a2 = bf2[t];
    #pragma unroll 4
    for (int k = 0; k < FC1_DIM; ++k) a2 += sh1[k] * Wf2[k * FC2_DIM + t];
    sh2[t] = a2;
  }
  __syncthreads();

  if (t == 0) {
    float mx = sh2[0];
    for (int j = 1; j < FC2_DIM; ++j) mx = fmaxf(mx, sh2[j]);
    float s = 0.0f;
    for (int j = 0; j < FC2_DIM; ++j) s += __expf(sh2[j] - mx);
    red[0] = mx;
    red[1] = s;
  }
  __syncthreads();

  if (t < FC2_DIM)
    out[(size_t)gid * FC2_DIM + t] = __expf(sh2[t] - red[0]) / red[1];
}

// ---------------------------------------------------------------------------
extern "C" void kernel_launch(void* const* d_in, const int* in_sizes, int n_in,
                              void* d_out, int out_size, void* d_ws, size_t ws_size,
                              hipStream_t stream) {
  (void)in_sizes; (void)n_in; (void)out_size; (void)ws_size;
  const float* node_attr = (const float*)d_in[0];
  const float* Wc        = (const float*)d_in[1];   // [3,128,128]
  const float* bc        = (const float*)d_in[2];   // [3,128]
  const float* Wf1       = (const float*)d_in[3];   // [128,64]
  const float* bf1       = (const float*)d_in[4];
  const float* Wf2       = (const float*)d_in[5];   // [64,10]
  const float* bf2       = (const float*)d_in[6];
  const int*   src       = (const int*)d_in[7];
  const int*   dst       = (const int*)d_in[8];
  const int*   batching  = (const int*)d_in[9];
  float* out = (float*)d_out;

  // Workspace layout (fp32): agg | xA | xB | g
  float* agg  = (float*)d_ws;
  float* xA   = agg + (size_t)N_NODES * D;
  float* xB   = xA  + (size_t)N_NODES * D;
  float* gbuf = xB  + (size_t)N_NODES * D;

  const int NF = N_NODES * D;                       // 12.8M
  const dim3 blk(256);
  const int nRowTiles  = N_NODES / 16;              // 6250
  const int edgeBlocks = (N_EDGES + 7) / 8;         // 1 wave per edge
  const int gemmBlocks = (nRowTiles + 7) / 8;       // 8 row tiles per block
  const int poolBlocks = (NF + 255) / 256;

  const float* xin = node_attr;
  float* bufs[3] = { xA, xB, xA };
  for (int layer = 0; layer < 3; ++layer) {
    zero_f32<<<2048, blk, 0, stream>>>(agg, NF);
    edge_scatter<<<edgeBlocks, blk, 0, stream>>>(xin, src, dst, agg, N_EDGES);
    conv_gemm_relu<<<gemmBlocks, blk, 0, stream>>>(
        agg, Wc + (size_t)layer * D * D, bc + (size_t)layer * D,
        bufs[layer], nRowTiles);
    xin = bufs[layer];
  }

  zero_f32<<<256, blk, 0, stream>>>(gbuf, N_GRAPHS * D);
  pool_scatter<<<poolBlocks, blk, 0, stream>>>(xin, batching, gbuf);
  head_kernel<<<N_GRAPHS, dim3(64), 0, stream>>>(gbuf, Wf1, bf1, Wf2, bf2, out);
}